// ChairMLP_25314537243128
// MI455X (gfx1250) — compile-verified
//
#include <hip/hip_runtime.h>

// ChairMLP forward linearization for gfx1250 (MI455X).
// Store-bandwidth-bound (~316MB outputs, > 192MB L2): all streamed outputs use
// non-temporal stores. Matmul chain uses V_WMMA_F32_16X16X4_F32 (fp32-exact).

typedef __attribute__((ext_vector_type(2))) float v2f;
typedef __attribute__((ext_vector_type(8))) float v8f;

#define NPTS   65536
#define HD     60
#define THREADS 128
#define WAVES   4
#define PBSTR   65              // padded row stride (floats): point-stride 260 == 4 mod 64 banks
#define PB_WAVE (64 * PBSTR)    // 16 pts * 4 rows
#define WTILE   3840            // 15 k-tiles * 64 n * 4 kk floats per layer

#define NTST(p, v) __builtin_nontemporal_store((v), (p))

__global__ __launch_bounds__(THREADS) void chairmlp_kernel(
    const float* __restrict__ x,
    const float* __restrict__ w1, const float* __restrict__ b1,
    const float* __restrict__ w2, const float* __restrict__ b2,
    const float* __restrict__ w3, const float* __restrict__ b3,
    const float* __restrict__ w4, const float* __restrict__ b4,
    const float* __restrict__ w5, const float* __restrict__ b5,
    float* __restrict__ out)
{
    __shared__ float Wsh[3 * WTILE];   // W2/3/4 in WMMA B-operand layout
    __shared__ float bsh[3 * 64];
    __shared__ float w5sh[64];
    __shared__ float PB[WAVES * PB_WAVE];  // per-wave [P;B] (64 rows x 60 cols, stride 65)

    const int tid  = threadIdx.x;
    const int lane = tid & 31;
    const int wid  = tid >> 5;
    const int lo   = lane & 15;
    const int hi   = lane >> 4;

    // ---- stage weights into LDS: Wsh[layer][kt][n][kk] = W[n][kt*4+kk] (B[k][n]=W^T) ----
    const float* Ws[3] = { w2, w3, w4 };
    const float* bs[3] = { b2, b3, b4 };
    for (int layer = 0; layer < 3; ++layer) {
        const float* W = Ws[layer];
        for (int idx = tid; idx < WTILE; idx += THREADS) {
            int kt = idx >> 8;
            int n  = (idx >> 2) & 63;
            int kk = idx & 3;
            int k  = kt * 4 + kk;            // always < 60
            Wsh[layer * WTILE + idx] = (n < HD) ? W[n * HD + k] : 0.f;
        }
        for (int idx = tid; idx < 64; idx += THREADS)
            bsh[layer * 64 + idx] = (idx < HD) ? bs[layer][idx] : 0.f;
    }
    for (int idx = tid; idx < 64; idx += THREADS)
        w5sh[idx] = (idx < HD) ? w5[idx] : 0.f;
    __syncthreads();

    const size_t base = (size_t)(blockIdx.x * WAVES + wid) * 16;  // 16 points per wave
    float* pb = &PB[wid * PB_WAVE];

    float* Sout = out + (size_t)NPTS;                  // [N,4,60]
    float* zso  = out + (size_t)NPTS * 1201;           // [N,4,1]

    // per-lane x for the 8 points this half-wave sees in C/D vgpr slots
    float xl[8][3];
#pragma unroll
    for (int v = 0; v < 8; ++v) {
        size_t p = base + v + 8 * hi;
        xl[v][0] = x[p * 3 + 0];
        xl[v][1] = x[p * 3 + 1];
        xl[v][2] = x[p * 3 + 2];
    }

    // ---- layer 1 (tiny, scalar): z1 = x.w1^T + b1; PB1 = [w1^T;b1] * s1 ----
    {
        float* alk1 = out + (size_t)NPTS * 241;
#pragma unroll
        for (int v = 0; v < 8; ++v) {
            int p = v + 8 * hi;
            size_t n = base + p;
#pragma unroll
            for (int nt = 0; nt < 4; ++nt) {
                int j = nt * 16 + lo;
                if (j < HD) {
                    float c0 = w1[j * 3 + 0], c1 = w1[j * 3 + 1], c2 = w1[j * 3 + 2];
                    float bb = b1[j];
                    float z = xl[v][0] * c0 + xl[v][1] * c1 + xl[v][2] * c2 + bb;
                    float s = (z > 0.f) ? 1.f : 0.f;
                    NTST(&Sout[n * 240 + j], s);
                    NTST(&alk1[n * 240 +   0 + j], c0);
                    NTST(&alk1[n * 240 +  60 + j], c1);
                    NTST(&alk1[n * 240 + 120 + j], c2);
                    NTST(&alk1[n * 240 + 180 + j], bb);
                    pb[(p * 4 + 0) * PBSTR + j] = c0 * s;
                    pb[(p * 4 + 1) * PBSTR + j] = c1 * s;
                    pb[(p * 4 + 2) * PBSTR + j] = c2 * s;
                    pb[(p * 4 + 3) * PBSTR + j] = bb * s;
                }
            }
        }
    }
    __builtin_amdgcn_wave_barrier();

    // ---- layers 2..4: alk_k = PB @ Wk^T (+bias row 3), mask, recycle through LDS ----
    for (int L = 0; L < 3; ++L) {
        const float* WshL = &Wsh[L * WTILE];
        const float* bL   = &bsh[L * 64];
        float* alkk = out + (size_t)NPTS * (size_t)(1 + 240 * (L + 2));

        v8f acc[4][4];   // [nt][r], all 16 tiles of the 64x64 product live
#pragma unroll
        for (int nt = 0; nt < 4; ++nt)
#pragma unroll
            for (int r = 0; r < 4; ++r)
                acc[nt][r] = (v8f){0.f, 0.f, 0.f, 0.f, 0.f, 0.f, 0.f, 0.f};

        for (int kt = 0; kt < 15; ++kt) {
            // A tiles: M-tile = r, in-tile M = point = lo; K = kt*4 + q + 2*hi
            v2f a[4];
#pragma unroll
            for (int r = 0; r < 4; ++r) {
                int off = (lo * 4 + r) * PBSTR + kt * 4 + 2 * hi;
                a[r].x = pb[off];
                a[r].y = pb[off + 1];
            }
#pragma unroll
            for (int nt = 0; nt < 4; ++nt) {
                int boff = kt * 256 + (nt * 16 + lo) * 4 + 2 * hi;  // 8B aligned
                v2f b;
                b.x = WshL[boff];
                b.y = WshL[boff + 1];
#pragma unroll
                for (int r = 0; r < 4; ++r)
                    acc[nt][r] = __builtin_amdgcn_wmma_f32_16x16x4_f32(
                        false, a[r], false, b, (short)0, acc[nt][r], false, false);
            }
        }
        __builtin_amdgcn_wave_barrier();   // all A reads done before pb is overwritten

#pragma unroll
        for (int nt = 0; nt < 4; ++nt) {
            int j = nt * 16 + lo;
            float bkv = bL[j];
            float sv[8];
#pragma unroll
            for (int i = 0; i < 8; ++i) {
                float a3 = acc[nt][3][i] + bkv;      // bias on B-row
                acc[nt][3][i] = a3;
                float z = xl[i][0] * acc[nt][0][i] + xl[i][1] * acc[nt][1][i] +
                          xl[i][2] * acc[nt][2][i] + a3;
                sv[i] = (z > 0.f) ? 1.f : 0.f;
            }
            if (j < HD) {
#pragma unroll
                for (int i = 0; i < 8; ++i) {
                    size_t n = base + i + 8 * hi;
                    NTST(&Sout[n * 240 + (size_t)(L + 1) * 60 + j], sv[i]);
                    NTST(&alkk[n * 240 +   0 + j], acc[nt][0][i]);
                    NTST(&alkk[n * 240 +  60 + j], acc[nt][1][i]);
                    NTST(&alkk[n * 240 + 120 + j], acc[nt][2][i]);
                    NTST(&alkk[n * 240 + 180 + j], acc[nt][3][i]);
                }
            }
#pragma unroll
            for (int i = 0; i < 8; ++i) {
                int p = i + 8 * hi;
                pb[(p * 4 + 0) * PBSTR + j] = acc[nt][0][i] * sv[i];
                pb[(p * 4 + 1) * PBSTR + j] = acc[nt][1][i] * sv[i];
                pb[(p * 4 + 2) * PBSTR + j] = acc[nt][2][i] * sv[i];
                pb[(p * 4 + 3) * PBSTR + j] = acc[nt][3][i] * sv[i];
            }
        }
        __builtin_amdgcn_wave_barrier();
    }

    // ---- layer 5: zs[r] = PB4[p][r] . w5 (+b5 on r=3); out = x.zs_w + zs_b ----
    float zs[4];
#pragma unroll
    for (int r = 0; r < 4; ++r) {
        float partial = 0.f;
        for (int jj = 0; jj < 30; ++jj) {
            int j = hi * 30 + jj;
            partial += pb[(lo * 4 + r) * PBSTR + j] * w5sh[j];
        }
        partial += __shfl_xor(partial, 16, 32);   // pair halves of the 60-dot
        zs[r] = partial;
    }
    zs[3] += b5[0];
    if (hi == 0) {
        size_t n = base + lo;
        NTST(&zso[n * 4 + 0], zs[0]);
        NTST(&zso[n * 4 + 1], zs[1]);
        NTST(&zso[n * 4 + 2], zs[2]);
        NTST(&zso[n * 4 + 3], zs[3]);
        float x0 = x[n * 3 + 0], x1 = x[n * 3 + 1], x2 = x[n * 3 + 2];
        NTST(&out[n], x0 * zs[0] + x1 * zs[1] + x2 * zs[2] + zs[3]);
    }
}

extern "C" void kernel_launch(void* const* d_in, const int* in_sizes, int n_in,
                              void* d_out, int out_size, void* d_ws, size_t ws_size,
                              hipStream_t stream) {
    (void)in_sizes; (void)n_in; (void)out_size; (void)d_ws; (void)ws_size;
    const float* x  = (const float*)d_in[0];
    const float* w1 = (const float*)d_in[1];
    const float* b1 = (const float*)d_in[2];
    const float* w2 = (const float*)d_in[3];
    const float* b2 = (const float*)d_in[4];
    const float* w3 = (const float*)d_in[5];
    const float* b3 = (const float*)d_in[6];
    const float* w4 = (const float*)d_in[7];
    const float* b4 = (const float*)d_in[8];
    const float* w5 = (const float*)d_in[9];
    const float* b5 = (const float*)d_in[10];
    float* out = (float*)d_out;

    const int blocks = NPTS / (WAVES * 16);   // 1024 blocks, 128 threads, 4 waves
    hipLaunchKernelGGL(chairmlp_kernel, dim3(blocks), dim3(THREADS), 0, stream,
                       x, w1, b1, w2, b2, w3, b3, w4, b4, w5, b5, out);
}